// VectorQuantizer_36283883717378
// MI455X (gfx1250) — compile-verified
//
#include <hip/hip_runtime.h>

// CDNA5 / gfx1250 VQ nearest-codebook kernel using V_WMMA_F32_16X16X4_F32.
//
// score[m][k] = ||w_k||^2 - 2 * x_m . w_k    (||x||^2 dropped: constant per row)
// WMMA computes D = A*B + C with A = -2*X tile, C initialized to ||w_k||^2.
// FOUR independent accumulator chains per iteration: the backend scheduler
// sinks each LDS B-fragment load to just before its consumer; with 4 chains
// interleaved, every load gets ~6 covering WMMAs from the other chains.
// Codebook staged global->LDS with gfx1250 async-LDS loads (ASYNCcnt).

typedef __attribute__((ext_vector_type(2))) float v2f;
typedef __attribute__((ext_vector_type(8))) float v8f;
typedef __attribute__((ext_vector_type(4))) int   v4i;

typedef __attribute__((address_space(1))) v4i* gv4i_ptr;   // global
typedef __attribute__((address_space(3))) v4i* lv4i_ptr;   // LDS

#define NUM_CODES 1024
#define DIM 64
#define ROWS_PER_WAVE 16
#define WAVES_PER_BLOCK 8
#define ROWS_PER_BLOCK (ROWS_PER_WAVE * WAVES_PER_BLOCK)   // 128

#if defined(__has_builtin)
#if __has_builtin(__builtin_amdgcn_global_load_async_to_lds_b128)
#define VQ_HAVE_ASYNC_LDS 1
#endif
#if __has_builtin(__builtin_amdgcn_s_wait_asynccnt)
#define VQ_HAVE_WAIT_ASYNC 1
#endif
#endif

__global__ void __launch_bounds__(256)
vq_wmma_kernel(const float* __restrict__ x,
               const float* __restrict__ w,
               float* __restrict__ out)
{
    // dynamic LDS: 1024*64 floats codebook + 1024 floats of ||w||^2  (260 KB)
    extern __shared__ float smem[];
    float* ldsW = smem;                       // [NUM_CODES * DIM]
    float* ldsE = smem + NUM_CODES * DIM;     // [NUM_CODES]

    const int tid  = threadIdx.x;             // 0..255
    const int lane = tid & 31;                // wave32
    const int wave = tid >> 5;                // 0..7
    const int half = lane >> 4;               // K-group / M-group selector
    const int l16  = lane & 15;

    // ---- Stage codebook into LDS ------------------------------------------
    {
        const float4* src = (const float4*)w;
        float4*       dst = (float4*)ldsW;
#if VQ_HAVE_ASYNC_LDS
        // gfx1250 async copy: global -> LDS without VGPR round-trip (ASYNCcnt)
        for (int i = tid; i < NUM_CODES * DIM / 4; i += 256) {
            __builtin_amdgcn_global_load_async_to_lds_b128(
                (gv4i_ptr)(src + i),
                (lv4i_ptr)(dst + i),
                /*offset=*/0, /*cpol=*/0);
        }
#if VQ_HAVE_WAIT_ASYNC
        __builtin_amdgcn_s_wait_asynccnt(0);
#else
        asm volatile("s_wait_asynccnt 0" ::: "memory");
#endif
#else
        #pragma unroll 4
        for (int i = tid; i < NUM_CODES * DIM / 4; i += 256) dst[i] = src[i];
#endif
    }
    __syncthreads();

    // ---- Per-code squared norms -------------------------------------------
    for (int c = tid; c < NUM_CODES; c += 256) {
        const float* row = ldsW + c * DIM;
        float s = 0.f;
        #pragma unroll
        for (int d = 0; d < DIM; ++d) s += row[d] * row[d];
        ldsE[c] = s;
    }
    __syncthreads();

    // ---- A fragments: 16 rows of X, pre-scaled by -2 ----------------------
    // ISA 16x4 f32 A layout: lane L<16 holds (K=0,K=1) in VGPR(0,1) for M=L;
    // lanes 16..31 hold (K=2,K=3). -> contiguous float2 at  4*kc + 2*half.
    const long rowBase = (long)blockIdx.x * ROWS_PER_BLOCK + (long)wave * ROWS_PER_WAVE;
    v2f a[16];
    {
        const float* xrow = x + (rowBase + l16) * DIM + 2 * half;
        #pragma unroll
        for (int kc = 0; kc < 16; ++kc) {
            v2f t = *(const v2f*)(xrow + 4 * kc);
            a[kc] = t * -2.0f;
        }
    }

    // Running argmin per C slot.  C layout: VGPR i, lanes 0-15 -> M=i,
    // lanes 16-31 -> M=i+8; N = l16 within the current 16-code block.
    float bestV[8];
    int   bestI[8];
    #pragma unroll
    for (int i = 0; i < 8; ++i) { bestV[i] = __builtin_inff(); bestI[i] = 0; }

    // Four 16-code blocks per iteration -> four independent WMMA chains.
    for (int quad = 0; quad < NUM_CODES / 64; ++quad) {
        const int n0 = quad * 64;
        const int n1 = n0 + 16;
        const int n2 = n0 + 32;
        const int n3 = n0 + 48;

        const float e0 = ldsE[n0 + l16];
        const float e1 = ldsE[n1 + l16];
        const float e2 = ldsE[n2 + l16];
        const float e3 = ldsE[n3 + l16];
        v8f c0 = { e0, e0, e0, e0, e0, e0, e0, e0 };
        v8f c1 = { e1, e1, e1, e1, e1, e1, e1, e1 };
        v8f c2 = { e2, e2, e2, e2, e2, e2, e2, e2 };
        v8f c3 = { e3, e3, e3, e3, e3, e3, e3, e3 };

        const float* wrow0 = ldsW + (n0 + l16) * DIM + 2 * half;
        const float* wrow1 = ldsW + (n1 + l16) * DIM + 2 * half;
        const float* wrow2 = ldsW + (n2 + l16) * DIM + 2 * half;
        const float* wrow3 = ldsW + (n3 + l16) * DIM + 2 * half;

        #pragma unroll
        for (int kc = 0; kc < 16; ++kc) {
            const v2f b0 = *(const v2f*)(wrow0 + 4 * kc);
            const v2f b1 = *(const v2f*)(wrow1 + 4 * kc);
            const v2f b2 = *(const v2f*)(wrow2 + 4 * kc);
            const v2f b3 = *(const v2f*)(wrow3 + 4 * kc);
            c0 = __builtin_amdgcn_wmma_f32_16x16x4_f32(
                     false, a[kc], false, b0, (short)0, c0, false, false);
            c1 = __builtin_amdgcn_wmma_f32_16x16x4_f32(
                     false, a[kc], false, b1, (short)0, c1, false, false);
            c2 = __builtin_amdgcn_wmma_f32_16x16x4_f32(
                     false, a[kc], false, b2, (short)0, c2, false, false);
            c3 = __builtin_amdgcn_wmma_f32_16x16x4_f32(
                     false, a[kc], false, b3, (short)0, c3, false, false);
        }

        // Ascending code order + strict '<' keeps the first (lowest) index on
        // ties, matching jnp.argmin.  n0 < n1 < n2 < n3 checked in order.
        #pragma unroll
        for (int i = 0; i < 8; ++i) {
            const float v0 = c0[i];
            if (v0 < bestV[i]) { bestV[i] = v0; bestI[i] = n0 + l16; }
            const float v1 = c1[i];
            if (v1 < bestV[i]) { bestV[i] = v1; bestI[i] = n1 + l16; }
            const float v2 = c2[i];
            if (v2 < bestV[i]) { bestV[i] = v2; bestI[i] = n2 + l16; }
            const float v3 = c3[i];
            if (v3 < bestV[i]) { bestV[i] = v3; bestI[i] = n3 + l16; }
        }
    }

    // ---- Cross-lane argmin within each 16-lane half (xor 1,2,4,8) ---------
    #pragma unroll
    for (int i = 0; i < 8; ++i) {
        float v   = bestV[i];
        int   idx = bestI[i];
        #pragma unroll
        for (int m = 1; m <= 8; m <<= 1) {
            const float ov = __shfl_xor(v, m, 32);
            const int   oi = __shfl_xor(idx, m, 32);
            if (ov < v || (ov == v && oi < idx)) { v = ov; idx = oi; }
        }
        bestV[i] = v;
        bestI[i] = idx;
    }

    // ---- Gather: row (rowBase + i + 8*half); 16 lanes x float4 = 64 floats
    #pragma unroll
    for (int i = 0; i < 8; ++i) {
        const long  row = rowBase + i + 8 * half;
        const float4 q  = *(const float4*)(ldsW + (long)bestI[i] * DIM + 4 * l16);
        *(float4*)(out + row * DIM + 4 * l16) = q;
    }
}

extern "C" void kernel_launch(void* const* d_in, const int* in_sizes, int n_in,
                              void* d_out, int out_size, void* d_ws, size_t ws_size,
                              hipStream_t stream) {
    const float* x   = (const float*)d_in[0];   // [32*4096, 64]
    const float* w   = (const float*)d_in[1];   // [1024, 64]
    float*       out = (float*)d_out;           // [32*4096, 64]

    const int M    = in_sizes[0] / DIM;         // 131072 (multiple of 128)
    const int grid = M / ROWS_PER_BLOCK;        // 1024 blocks

    const size_t shmem = (size_t)(NUM_CODES * DIM + NUM_CODES) * sizeof(float);

    vq_wmma_kernel<<<grid, 256, shmem, stream>>>(x, w, out);
}